// GCNLayer_57320633532847
// MI455X (gfx1250) — compile-verified
//
#include <hip/hip_runtime.h>
#include <cstdint>

typedef float v2f __attribute__((ext_vector_type(2)));
typedef float v8f __attribute__((ext_vector_type(8)));

#define NNODES 50000
#define NEDGES 625000
#define DIM    128

// ---------------------------------------------------------------------------
// Kernel 0: zero the accumulation buffers (d_out = sums, counts in ws).
// Re-run every launch so graph replays are deterministic.
// ---------------------------------------------------------------------------
__global__ void gcn_zero_kernel(float* __restrict__ out, float* __restrict__ counts) {
  int i = blockIdx.x * blockDim.x + threadIdx.x;
  if (i < NNODES * DIM) out[i] = 0.0f;
  if (i < NNODES)       counts[i] = 0.0f;
}

// ---------------------------------------------------------------------------
// Kernel 1: h = x @ W^T + b via V_WMMA_F32_16X16X4_F32 (full f32 precision).
// One wave32 computes one 16x16 tile of h. 3125 M-tiles x 8 N-tiles = 25000
// tiles, 4 waves / block -> 6250 blocks. No divergence: EXEC is all ones.
//
// A (16x4 f32): lane l holds row M = l%16; VGPR0/1 = K = 2*(l/16) + {0,1}.
// B (4x16 f32): lane l holds col N = l%16; VGPR0/1 = K = 2*(l/16) + {0,1},
//               element B[k][n] = W[n][k]  (h = x * W^T).
// C/D (16x16 f32): VGPR r, lane l -> M = r + 8*(l/16), N = l%16.
// ---------------------------------------------------------------------------
__global__ __launch_bounds__(128) void gcn_gemm_wmma_kernel(
    const float* __restrict__ x, const float* __restrict__ W,
    const float* __restrict__ bias, float* __restrict__ h) {
  const int wave = threadIdx.x >> 5;
  const int lane = threadIdx.x & 31;
  const int tile = blockIdx.x * 4 + wave;     // 0..24999
  const int mt = tile >> 3;                   // 0..3124
  const int nt = tile & 7;                    // 0..7

  const int mn    = lane & 15;                // M (for A) / N (for B)
  const int khalf = (lane >> 4) << 1;         // 0 or 2

  const float* __restrict__ xrow = x + (size_t)(mt * 16 + mn) * DIM;
  const float* __restrict__ wrow = W + (size_t)(nt * 16 + mn) * DIM;

  v8f c0 = {};
  v8f c1 = {};
  #pragma unroll
  for (int k = 0; k < DIM; k += 8) {
    v2f a0, b0, a1, b1;
    a0.x = xrow[k + khalf];     a0.y = xrow[k + khalf + 1];
    b0.x = wrow[k + khalf];     b0.y = wrow[k + khalf + 1];
    a1.x = xrow[k + 4 + khalf]; a1.y = xrow[k + 4 + khalf + 1];
    b1.x = wrow[k + 4 + khalf]; b1.y = wrow[k + 4 + khalf + 1];
    // (neg_a, A, neg_b, B, c_mod, C, reuse_a, reuse_b)
    c0 = __builtin_amdgcn_wmma_f32_16x16x4_f32(false, a0, false, b0,
                                               (short)0, c0, false, false);
    c1 = __builtin_amdgcn_wmma_f32_16x16x4_f32(false, a1, false, b1,
                                               (short)0, c1, false, false);
  }

  const int ncol = nt * 16 + mn;
  const float bv = bias[ncol];
  const int mbase = mt * 16 + ((lane >> 4) << 3);
  #pragma unroll
  for (int r = 0; r < 8; ++r) {
    h[(size_t)(mbase + r) * DIM + ncol] = c0[r] + c1[r] + bv;
  }
}

// ---------------------------------------------------------------------------
// Kernel 2: edge-parallel scatter. One wave per edge; lane j adds elements
// j, j+32, j+64, j+96 of h[src] into sums[dst] with global_atomic_add_f32.
// h and sums are L2-resident (25.6 MB each vs 192 MB L2).
// ---------------------------------------------------------------------------
__global__ __launch_bounds__(256) void gcn_scatter_kernel(
    const float* __restrict__ h, const long long* __restrict__ ei,
    float* __restrict__ sums, float* __restrict__ counts) {
  const int e    = (int)((blockIdx.x * (unsigned)blockDim.x + threadIdx.x) >> 5);
  const int lane = threadIdx.x & 31;
  if (e >= NEDGES) return;

  const long long src = ei[e];           // edge_index[0][e]
  const long long dst = ei[NEDGES + e];  // edge_index[1][e]

  const float* __restrict__ hs = h + (size_t)src * DIM;
  float* __restrict__ sd = sums + (size_t)dst * DIM;

  #pragma unroll
  for (int j = 0; j < 4; ++j) {
    const int idx = lane + 32 * j;
    atomicAdd(&sd[idx], hs[idx]);
  }
  if (lane == 0) atomicAdd(&counts[dst], 1.0f);
}

// ---------------------------------------------------------------------------
// Kernel 3: mean (guard empty segments) + ReLU, in place on d_out.
// ---------------------------------------------------------------------------
__global__ void gcn_finalize_kernel(float* __restrict__ out,
                                    const float* __restrict__ counts) {
  const int i = blockIdx.x * blockDim.x + threadIdx.x;
  if (i >= NNODES * DIM) return;
  const float cnt = counts[i >> 7];
  const float v = out[i] / fmaxf(cnt, 1.0f);
  out[i] = fmaxf(v, 0.0f);
}

extern "C" void kernel_launch(void* const* d_in, const int* in_sizes, int n_in,
                              void* d_out, int out_size, void* d_ws, size_t ws_size,
                              hipStream_t stream) {
  const float*     x    = (const float*)d_in[0];
  const long long* ei   = (const long long*)d_in[1];  // int64 edge_index [2, E]
  const float*     W    = (const float*)d_in[2];
  const float*     bias = (const float*)d_in[3];
  float* out = (float*)d_out;

  // workspace: h [NNODES*DIM floats] | counts [NNODES floats]
  float* h      = (float*)d_ws;
  float* counts = h + (size_t)NNODES * DIM;

  // zero sums + counts (every call, for deterministic graph replay)
  {
    int n = NNODES * DIM;
    gcn_zero_kernel<<<(n + 255) / 256, 256, 0, stream>>>(out, counts);
  }
  // GEMM: 25000 tiles, 4 waves (128 threads) per block
  gcn_gemm_wmma_kernel<<<6250, 128, 0, stream>>>(x, W, bias, h);
  // scatter: 1 wave per edge, 8 waves per block
  gcn_scatter_kernel<<<(NEDGES + 7) / 8, 256, 0, stream>>>(h, ei, out, counts);
  // mean + relu
  {
    int n = NNODES * DIM;
    gcn_finalize_kernel<<<(n + 255) / 256, 256, 0, stream>>>(out, counts);
  }
}